// SOLOV2ClsPredsDecoder_63891933495891
// MI455X (gfx1250) — compile-verified
//
#include <hip/hip_runtime.h>
#include <hip/hip_bf16.h>

// SOLOv2 cls-preds decoder for gfx1250 (MI455X).
// Memory-bound: ~158.6 MB traffic -> ~6.8 us at 23.3 TB/s. No matmul -> no WMMA.
// CDNA5 paths used: GLOBAL_LOAD_ASYNC_TO_LDS_B128 (ASYNCcnt) + s_wait_asynccnt,
// LDS transpose CHW->HWC, b128 LDS sigmoid pass (1 exp per input element),
// vectorized b128 LDS neighbor reads, coalesced 128B wave stores.

#define NUM_CLASSES 80
#define THREADS     256
#define TILES_PER_B 26      // 10 + 9 + 4 + 2 + 1

__device__ __forceinline__ float sigf(float x) {
    return 1.0f / (1.0f + __expf(-x));
}

__global__ __launch_bounds__(THREADS)
void solov2_cls_decode_kernel(const float* __restrict__ in0,
                              const float* __restrict__ in1,
                              const float* __restrict__ in2,
                              const float* __restrict__ in3,
                              const float* __restrict__ in4,
                              float* __restrict__ out) {
    // 80 planes * 204 floats = 65,280 bytes of LDS (max over levels)
    __shared__ __align__(16) float lds[NUM_CLASSES * 204];

    const int bid = blockIdx.x;
    const int b   = bid / TILES_PER_B;
    const int t   = bid % TILES_PER_B;

    // Per-level constants (S, R rows/tile, cell base in concat, LDS plane stride).
    // planeStride: multiple of 4 (b128 align), == 4 mod 8 (2-way bank conflicts max),
    // >= (R+1)*S. LDS <= 64KB for all levels.
    int S, R, rt, cb, ps;
    const float* in;
    if (t < 10)      { S = 40; R = 4;  rt = t;      cb = 0;    ps = 204; in = in0; }
    else if (t < 19) { S = 36; R = 4;  rt = t - 10; cb = 1600; ps = 180; in = in1; }
    else if (t < 23) { S = 24; R = 7;  rt = t - 19; cb = 2896; ps = 196; in = in2; }
    else if (t < 25) { S = 16; R = 8;  rt = t - 23; cb = 3472; ps = 148; in = in3; }
    else             { S = 12; R = 12; rt = 0;      cb = 3728; ps = 156; in = in4; }

    const int i0        = rt * R;
    const int startRow  = (i0 == 0) ? 0 : (i0 - 1);        // halo row above (if any)
    const int endRow    = min(i0 + R, S);
    const int chunkRows = endRow - startRow;
    const int chunkFl   = chunkRows * S;                   // contiguous floats per channel
    const int perChanV  = chunkFl >> 2;                    // float4 chunks per channel
    const int totalV    = NUM_CLASSES * perChanV;

    const size_t planeElems = (size_t)S * (size_t)S;
    const float* inB = in + (size_t)b * NUM_CLASSES * planeElems + (size_t)startRow * S;

    // LDS byte address of array base: low 32 bits of the generic shared address
    // are the LDS offset (aperture encoding keeps SHARED_BASE in the high bits).
    const unsigned ldsBase = (unsigned)(size_t)(&lds[0]);

    // ---- Phase 1: async DMA global -> LDS, 16B granules, fully coalesced ----
    for (int v = threadIdx.x; v < totalV; v += THREADS) {
        const int c = v / perChanV;
        const int k = v - c * perChanV;
        const float* g = inB + (size_t)c * planeElems + ((size_t)k << 2);
        const unsigned la = ldsBase + ((unsigned)(c * ps + (k << 2)) << 2);
        asm volatile("global_load_async_to_lds_b128 %0, %1, off"
                     :: "v"(la), "v"(g)
                     : "memory");
    }
    asm volatile("s_wait_asynccnt 0" ::: "memory");
    __syncthreads();   // all waves' async writes now visible in LDS

    // ---- Phase 1.5: in-place sigmoid over LDS, b128, conflict-free ----
    // Converting the (unused) inter-plane gap floats is harmless.
    const int psTot = NUM_CLASSES * 204;   // full buffer; multiple of 4
    for (int u = (threadIdx.x << 2); u < psTot; u += (THREADS << 2)) {
        float4 v4 = *(const float4*)&lds[u];
        v4.x = sigf(v4.x);
        v4.y = sigf(v4.y);
        v4.z = sigf(v4.z);
        v4.w = sigf(v4.w);
        *(float4*)&lds[u] = v4;
    }
    __syncthreads();

    // ---- Phase 2: peak suppression + transpose; pure compares ----
    // Work item w -> (channel c = w % 80, cell-vector pv = w / 80).
    // Each thread handles 4 consecutive cells of one row (S % 4 == 0, so a
    // 4-aligned run never crosses a row): 2x ds_load_b128 + <=2 edge b32 reads.
    // Lane-consecutive c keeps every b32 store a coalesced 128B wave store.
    const int rowsOut = endRow - i0;
    const int cells   = rowsOut * S;
    const int nvec    = cells >> 2;
    const int totalW  = nvec * NUM_CLASSES;
    const int topOfs  = i0 - startRow;  // local row of global row gi is (gi - i0) + topOfs
    const float NEGINF = -__builtin_inff();

    float* outB = out + ((size_t)b * 3872 + (size_t)cb + (size_t)i0 * S) * NUM_CLASSES;

    for (int w = threadIdx.x; w < totalW; w += THREADS) {
        const int pv = w / NUM_CLASSES;
        const int c  = w - pv * NUM_CLASSES;
        const int p0 = pv << 2;             // first cell of the 4-run
        const int pr = p0 / S;              // row within tile
        const int j0 = p0 - pr * S;         // column (multiple of 4)
        const int gi = i0 + pr;             // global row
        const int li = pr + topOfs;         // local row in LDS plane
        const int base = c * ps + li * S + j0;

        const float4 ctr = *(const float4*)&lds[base];

        // left neighbors: shift of ctr, with -inf at the pad edge (j==0)
        const float lEdge = (j0 > 0) ? lds[base - 1] : NEGINF;

        float4 top;
        float  tlEdge;
        if (gi > 0) {
            top    = *(const float4*)&lds[base - S];
            tlEdge = (j0 > 0) ? lds[base - S - 1] : NEGINF;
        } else {
            top    = make_float4(NEGINF, NEGINF, NEGINF, NEGINF);
            tlEdge = NEGINF;
        }

        const float o0 = (ctr.x >= lEdge  && ctr.x >= top.x && ctr.x >= tlEdge) ? ctr.x : 0.0f;
        const float o1 = (ctr.y >= ctr.x  && ctr.y >= top.y && ctr.y >= top.x ) ? ctr.y : 0.0f;
        const float o2 = (ctr.z >= ctr.y  && ctr.z >= top.z && ctr.z >= top.y ) ? ctr.z : 0.0f;
        const float o3 = (ctr.w >= ctr.z  && ctr.w >= top.w && ctr.w >= top.z ) ? ctr.w : 0.0f;

        float* dst = outB + (size_t)p0 * NUM_CLASSES + c;
        dst[0 * NUM_CLASSES] = o0;
        dst[1 * NUM_CLASSES] = o1;
        dst[2 * NUM_CLASSES] = o2;
        dst[3 * NUM_CLASSES] = o3;
    }
}

extern "C" void kernel_launch(void* const* d_in, const int* in_sizes, int n_in,
                              void* d_out, int out_size, void* d_ws, size_t ws_size,
                              hipStream_t stream) {
    (void)in_sizes; (void)n_in; (void)out_size; (void)d_ws; (void)ws_size;
    const float* s0 = (const float*)d_in[0];
    const float* s1 = (const float*)d_in[1];
    const float* s2 = (const float*)d_in[2];
    const float* s3 = (const float*)d_in[3];
    const float* s4 = (const float*)d_in[4];
    float* out = (float*)d_out;

    const int nblocks = 64 * TILES_PER_B;   // 1664 workgroups
    solov2_cls_decode_kernel<<<nblocks, THREADS, 0, stream>>>(s0, s1, s2, s3, s4, out);
}